// HeatmapBatch_19129784336624
// MI455X (gfx1250) — compile-verified
//
#include <hip/hip_runtime.h>

#define B   64
#define KP  21
#define H   256
#define KS  9
#define PAD 4

typedef __attribute__((ext_vector_type(2))) float v2f;
typedef __attribute__((ext_vector_type(4))) float v4f;
typedef __attribute__((ext_vector_type(8))) float v8f;

// Pass 1: stream 352 MB of zeros with 128-bit non-temporal stores.
// Output (352 MB) exceeds L2 (192 MB) and is never re-read -> NT hint.
__global__ void heatmap_zero_fill(float* __restrict__ out, size_t n) {
    const size_t gid    = (size_t)blockIdx.x * blockDim.x + threadIdx.x;
    const size_t stride = (size_t)gridDim.x * blockDim.x;
    const size_t n4 = n >> 2;
    v4f z = {0.0f, 0.0f, 0.0f, 0.0f};
    v4f* __restrict__ out4 = (v4f*)out;
    for (size_t i = gid; i < n4; i += stride)
        __builtin_nontemporal_store(z, out4 + i);
    // safety tail (out_size is divisible by 4 here, so normally dead)
    for (size_t i = (n4 << 2) + gid; i < n; i += stride)
        out[i] = 0.0f;
}

// Pass 2: one wave32 per (b,k) plane. The 9x9 Gaussian patch is the rank-1
// outer product 10*g[m]*g[n], computed with a single V_WMMA_F32_16X16X4_F32:
//   A[m,k] = 10*g[m]  (replicated over all K, so result is K-layout agnostic)
//   B[k,n] = 0.25*g[n]
//   D[m,n] = 4 * (10*g[m]) * (0.25*g[n]) = 10*g[m]*g[n] = 10*kernel2d[m][n]
// g[m] recovered from the kernel diagonal: g[m] = sqrt(k2d[m][m]).
__global__ __launch_bounds__(32) void heatmap_patch_wmma(
    const int* __restrict__ x, const float* __restrict__ k2d,
    float* __restrict__ out) {
    const int p    = blockIdx.x;      // plane id = b*KP + k
    const int lane = threadIdx.x;     // 0..31, EXEC all ones (no branches pre-WMMA)
    const int r = x[2 * p + 0];       // scatter row
    const int c = x[2 * p + 1];       // scatter col

    // Per-lane Gaussian value; m and n both equal lane&15 so one value serves
    // both the A and B operands. All conditionals are selects (no EXEC change).
    const int   idx  = lane & 15;
    const int   ci   = (idx < KS) ? idx : (KS - 1);           // clamp (cndmask)
    const float diag = k2d[ci * (KS + 1)];                    // k2d[ci][ci] = g[ci]^2
    const float msk  = (idx < KS) ? 1.0f : 0.0f;              // cndmask
    const float g    = sqrtf(diag) * msk;                     // g[idx], 0 outside 9

    v2f a;  a[0] = 10.0f * g;  a[1] = 10.0f * g;    // A 16x4, replicated over K
    v2f bm; bm[0] = 0.25f * g; bm[1] = 0.25f * g;   // B 4x16, replicated over K
    v8f cc = {};
    v8f d = __builtin_amdgcn_wmma_f32_16x16x4_f32(
        /*neg_a=*/false, a, /*neg_b=*/false, bm,
        /*c_mod=*/(short)0, cc, /*reuse_a=*/false, /*reuse_b=*/false);

    // C/D layout: lanes 0-15: VGPR v -> (M=v,   N=lane)
    //             lanes 16-31: VGPR v -> (M=v+8, N=lane-16)
    const int n     = lane & 15;
    const int mbase = (lane >> 4) * 8;
    const int col   = c + n - PAD;
    float* __restrict__ plane = out + (size_t)p * H * H;
    if (n < KS && (unsigned)col < (unsigned)H) {
        #pragma unroll
        for (int v = 0; v < 8; ++v) {
            const int m   = mbase + v;
            const int row = r + m - PAD;
            // bounds clip == conv zero-padding edge behavior
            if (m < KS && (unsigned)row < (unsigned)H)
                plane[row * H + col] = d[v];
        }
    }
}

extern "C" void kernel_launch(void* const* d_in, const int* in_sizes, int n_in,
                              void* d_out, int out_size, void* d_ws, size_t ws_size,
                              hipStream_t stream) {
    const int*   x   = (const int*)d_in[0];     // [B, KP, 2] int32
    const float* k2d = (const float*)d_in[1];   // [KS, KS] float32
    float*       out = (float*)d_out;           // [B, KP, H, H] float32

    const size_t n = (size_t)out_size;          // 88,080,384 floats = 352 MB

    // Pass 1: bandwidth-bound zero fill (grid-stride, plenty of waves in flight)
    heatmap_zero_fill<<<20480, 256, 0, stream>>>(out, n);

    // Pass 2: one wave per plane; WMMA outer-product patch + clipped scatter
    heatmap_patch_wmma<<<B * KP, 32, 0, stream>>>(x, k2d, out);
}